// RNNDecoderModule_82798379532736
// MI455X (gfx1250) — compile-verified
//
#include <hip/hip_runtime.h>
#include <math.h>

// ---------------------------------------------------------------------------
// RNN decoder step (2-layer LSTM, batch=1) for gfx1250 / MI455X.
// Bandwidth-bound fp32 GEMV chain (~273 MB weights, read once; HBM floor
// ~11.7us at 23.3 TB/s; FLOPs negligible). GEMVs use V_WMMA_F32_16X16X4_F32:
//   A = 16 rows x 4 k of W, B = x chunk broadcast across all 16 columns,
//   so every column of D accumulates y[m] = sum_k W[m,k]*x[k].
// The two lane-halves of a wave stream DISJOINT contiguous k-ranges (the
// WMMA just sums its 4 k-products, so we pick which k each half carries);
// that makes each lane's W stream sequential -> b128 non-temporal loads,
// one 16B load per two WMMAs. Scalar (readfirstlane) loop bounds keep the
// K-loop uniform so it unrolls. Deterministic LDS reductions, no atomics.
// ---------------------------------------------------------------------------

typedef __attribute__((ext_vector_type(2))) float v2f;
typedef __attribute__((ext_vector_type(4))) float v4f;
typedef __attribute__((ext_vector_type(8))) float v8f;

#define EMB_E   1024
#define FOUR_E  4096
#define VOCAB   50257
#define NLAYER  2
#define GEMV_WAVES 4   // 128 threads / block (4 wave32)

__device__ __forceinline__ float sigmoidf_(float v) {
    return 1.0f / (1.0f + expf(-v));
}

// ------------------------- embedding + relu -------------------------------
__global__ void embed_relu_kernel(const int* __restrict__ token,
                                  const float* __restrict__ emb,
                                  float* __restrict__ x) {
    int i = blockIdx.x * blockDim.x + threadIdx.x;
    if (i < EMB_E) {
        float v = emb[(size_t)token[0] * EMB_E + i];
        x[i] = v > 0.0f ? v : 0.0f;
    }
}

// ------------------- 16-row GEMV tile via WMMA f32 ------------------------
// y[m] = W1[m,:].x1 (+ W2[m,:].x2) (+ b1[m]) (+ b2[m]) for m in [row0,row0+16)
// If W2 != null: waves 0-1 cover W1 (K/2 each), waves 2-3 cover W2.
// Else: all 4 waves split K on W1. Within a wave, lane-half 0 streams the
// first half of the wave's k-chunk, lane-half 1 the second half.
__global__ void gemv16_wmma_kernel(const float* __restrict__ W1,
                                   const float* __restrict__ x1,
                                   const float* __restrict__ W2,
                                   const float* __restrict__ x2,
                                   const float* __restrict__ b1,
                                   const float* __restrict__ b2,
                                   float* __restrict__ y,
                                   int M, int K) {
    const int wave = __builtin_amdgcn_readfirstlane(threadIdx.x >> 5); // scalar
    const int lane = threadIdx.x & 31;
    const int half = lane >> 4;     // which k-substream this lane carries
    const int mrow = lane & 15;
    const int row0 = blockIdx.x * 16;

    int m = row0 + mrow;
    if (m >= M) m = M - 1;          // clamp loads; stores guarded below

    const float* Wsel;
    const float* xsel;
    int kstart, klen;
    if (W2 != nullptr) {
        const int kc = K >> 1;      // 2 waves per matrix
        if (wave < 2) { Wsel = W1; xsel = x1; kstart = wave * kc; }
        else          { Wsel = W2; xsel = x2; kstart = (wave - 2) * kc; }
        klen = kc;
    } else {
        const int kc = K >> 2;      // 4 waves on W1
        Wsel = W1; xsel = x1;
        kstart = wave * kc; klen = kc;
    }

    const int hlen = klen >> 1;                 // k per lane-half (>=128, %16==0)
    const int hoff = kstart + half * hlen;
    const float* __restrict__ wp = Wsel + (size_t)m * K + hoff;
    const float* __restrict__ xp = xsel + hoff;

    v8f acc = {};
    #pragma unroll 4
    for (int p = 0; p < hlen; p += 4) {
        // Streaming, read-once weights: non-temporal b128 loads.
        v4f w  = __builtin_nontemporal_load((const v4f*)(wp + p));
        v4f xv = *(const v4f*)(xp + p);
        v2f a0 = {w.x, w.y}, b0 = {xv.x, xv.y};
        v2f a1 = {w.z, w.w}, b1v = {xv.z, xv.w};
        // 8 args: (neg_a, A, neg_b, B, c_mod, C, reuse_a, reuse_b)
        acc = __builtin_amdgcn_wmma_f32_16x16x4_f32(
            false, a0, false, b0, (short)0, acc, false, false);
        acc = __builtin_amdgcn_wmma_f32_16x16x4_f32(
            false, a1, false, b1v, (short)0, acc, false, false);
    }

    // C/D layout: VGPR j -> row row0+j on lanes 0-15, row row0+8+j on 16-31.
    // All columns identical -> read column 0 from lane 0 and lane 16.
    __shared__ float red[GEMV_WAVES][2][8];
    if (lane == 0 || lane == 16) {
        #pragma unroll
        for (int j = 0; j < 8; ++j) red[wave][half][j] = acc[j];
    }
    __syncthreads();

    if (threadIdx.x < 16) {
        const int t    = threadIdx.x;
        const int part = t >> 3;
        const int j    = t & 7;
        float s = 0.0f;
        #pragma unroll
        for (int w = 0; w < GEMV_WAVES; ++w) s += red[w][part][j];
        const int row = row0 + t;   // part*8 + j == t
        if (row < M) {
            if (b1) s += b1[row];
            if (b2) s += b2[row];
            y[row] = s;
        }
    }
}

// ------------------------- LSTM pointwise ---------------------------------
__global__ void lstm_pointwise_kernel(const float* __restrict__ gates,
                                      const float* __restrict__ c_in,
                                      float* __restrict__ h_out,
                                      float* __restrict__ c_out,
                                      float* __restrict__ x_next) {
    int t = blockIdx.x * blockDim.x + threadIdx.x;
    if (t < EMB_E) {
        float i = sigmoidf_(gates[t]);
        float f = sigmoidf_(gates[EMB_E + t]);
        float g = tanhf(gates[2 * EMB_E + t]);
        float o = sigmoidf_(gates[3 * EMB_E + t]);
        float c = f * c_in[t] + i * g;
        float h = o * tanhf(c);
        h_out[t]  = h;
        c_out[t]  = c;
        x_next[t] = h;
    }
}

// ------------------------- log-softmax (1 block) --------------------------
__global__ void log_softmax_kernel(const float* __restrict__ logits,
                                   float* __restrict__ out, int n) {
    __shared__ float sdata[1024];
    const int tid = threadIdx.x;
    const int nt  = blockDim.x;

    float m = -INFINITY;
    for (int i = tid; i < n; i += nt) m = fmaxf(m, logits[i]);
    sdata[tid] = m;
    __syncthreads();
    for (int s = nt >> 1; s > 0; s >>= 1) {
        if (tid < s) sdata[tid] = fmaxf(sdata[tid], sdata[tid + s]);
        __syncthreads();
    }
    const float gmax = sdata[0];
    __syncthreads();

    float sum = 0.0f;
    for (int i = tid; i < n; i += nt) sum += expf(logits[i] - gmax);
    sdata[tid] = sum;
    __syncthreads();
    for (int s = nt >> 1; s > 0; s >>= 1) {
        if (tid < s) sdata[tid] += sdata[tid + s];
        __syncthreads();
    }
    const float lse = gmax + logf(sdata[0]);

    for (int i = tid; i < n; i += nt) out[i] = logits[i] - lse;
}

// --------------------------------------------------------------------------
extern "C" void kernel_launch(void* const* d_in, const int* in_sizes, int n_in,
                              void* d_out, int out_size, void* d_ws, size_t ws_size,
                              hipStream_t stream) {
    (void)in_sizes; (void)n_in; (void)out_size; (void)ws_size;

    const int*   token = (const int*)d_in[0];
    const float* h0    = (const float*)d_in[1];
    const float* c0    = (const float*)d_in[2];
    const float* emb   = (const float*)d_in[3];
    const float* w_ih  = (const float*)d_in[4];
    const float* w_hh  = (const float*)d_in[5];
    const float* b_ih  = (const float*)d_in[6];
    const float* b_hh  = (const float*)d_in[7];
    const float* out_w = (const float*)d_in[8];
    const float* out_b = (const float*)d_in[9];
    float* out = (float*)d_out;

    float* ws     = (float*)d_ws;
    float* x      = ws;                      // E floats (current layer input)
    float* gates  = ws + EMB_E;              // 4E floats
    float* logits = ws + EMB_E + FOUR_E;     // VOCAB floats

    float* hn = out + VOCAB;                 // (L,1,E)
    float* cn = out + VOCAB + NLAYER * EMB_E;// (L,1,E)

    embed_relu_kernel<<<(EMB_E + 255) / 256, 256, 0, stream>>>(token, emb, x);

    for (int l = 0; l < NLAYER; ++l) {
        gemv16_wmma_kernel<<<FOUR_E / 16, 32 * GEMV_WAVES, 0, stream>>>(
            w_ih + (size_t)l * FOUR_E * EMB_E, x,
            w_hh + (size_t)l * FOUR_E * EMB_E, h0 + (size_t)l * EMB_E,
            b_ih + (size_t)l * FOUR_E, b_hh + (size_t)l * FOUR_E,
            gates, FOUR_E, EMB_E);
        lstm_pointwise_kernel<<<(EMB_E + 255) / 256, 256, 0, stream>>>(
            gates, c0 + (size_t)l * EMB_E,
            hn + (size_t)l * EMB_E, cn + (size_t)l * EMB_E, x);
    }

    gemv16_wmma_kernel<<<(VOCAB + 15) / 16, 32 * GEMV_WAVES, 0, stream>>>(
        out_w, x, nullptr, nullptr, out_b, nullptr, logits, VOCAB, EMB_E);

    log_softmax_kernel<<<1, 1024, 0, stream>>>(logits, out, VOCAB);
}